// NMTGrounding_10170482557436
// MI455X (gfx1250) — compile-verified
//
#include <hip/hip_runtime.h>
#include <math.h>

// ---------------------------------------------------------------------------
// NMT grounding pipeline for MI455X (gfx1250, wave32, WMMA + TDM).
// Dense GEMMs: V_WMMA_F32_16X16X4_F32 (fp32 kept: recurrence is L2-BW bound,
// ~6.5 flop/byte, so bf16's higher K-rate would not be realized, and fp32
// avoids compounding recurrence error). Serial LSTM = 64 graph-captured step
// launches with ping-pong h buffers; h_in staged into LDS once per WG via
// TENSOR_LOAD_TO_LDS (TDM, TENSORcnt). Tail scoring fused into one kernel.
// ---------------------------------------------------------------------------

typedef __attribute__((ext_vector_type(2))) float v2f;
typedef __attribute__((ext_vector_type(8))) float v8f;
typedef __attribute__((ext_vector_type(4))) unsigned int u32x4;
typedef __attribute__((ext_vector_type(8))) unsigned int u32x8;

#define Bz 16
#define Tz 64
#define Rz 100
#define EMBED 300
#define RNN 1024
#define SCd 256

__device__ __forceinline__ float sigf(float x) { return 1.0f / (1.0f + expf(-x)); }

__device__ __forceinline__ float wred(float v) {
#pragma unroll
    for (int m = 16; m >= 1; m >>= 1) v += __shfl_xor(v, m, 32);
    return v;   // all lanes hold the sum (xor butterfly)
}

__global__ void zero_kernel(float* p, int n) {
    int i = blockIdx.x * blockDim.x + threadIdx.x;
    if (i < n) p[i] = 0.f;
}

// ---------------------------------------------------------------------------
// Generic C[M,N] = A[M,K] @ W[K,N] (+ bias[N]); row major; one wave per block.
// Wave computes one 16-row M tile and NT 16-col N tiles (A-frag reused NT x).
// Fragment layouts per CDNA5 ISA 7.12.2:
//   A 16x4 : lane l(0..15) hi(0/1): v0=A[m0+l][ka], v1=A[m0+l][ka+1], ka=kk+2*hi
//   B 4x16 : v0=W[ka][n0+l], v1=W[ka+1][n0+l]
//   C/D    : vgpr v: rows m0+v (lanes0-15) / m0+v+8 (lanes16-31), col n0+l
// Requires M%16==0, K%4==0. NTAIL predicates the N edge (N=300 case).
// ---------------------------------------------------------------------------
template <int NT, bool NTAIL>
__global__ void gemm_wmma(const float* __restrict__ A, const float* __restrict__ W,
                          const float* __restrict__ bias, float* __restrict__ C,
                          int M, int N, int K) {
    int lane = threadIdx.x & 31;
    int l = lane & 15, hi = lane >> 4;
    int m0 = blockIdx.x * 16;
    int n0 = blockIdx.y * (16 * NT);

    v8f acc[NT];
#pragma unroll
    for (int j = 0; j < NT; j++)
#pragma unroll
        for (int v = 0; v < 8; v++) acc[j][v] = 0.f;

    const float* Arow = A + (size_t)(m0 + l) * K;
    for (int kk = 0; kk < K; kk += 4) {
        int ka = kk + 2 * hi;
        v2f a = *(const v2f*)(Arow + ka);          // 8B aligned: ka even, K%4==0
        const float* Wr = W + (size_t)ka * N;
#pragma unroll
        for (int j = 0; j < NT; j++) {
            int col = n0 + j * 16 + l;
            v2f b;
            if (NTAIL && col >= N) { b.x = 0.f; b.y = 0.f; }
            else { b.x = Wr[col]; b.y = Wr[(size_t)N + col]; }
            acc[j] = __builtin_amdgcn_wmma_f32_16x16x4_f32(
                false, a, false, b, (short)0, acc[j], false, false);
        }
    }

#pragma unroll
    for (int j = 0; j < NT; j++) {
        int col = n0 + j * 16 + l;
        if (NTAIL && col >= N) continue;
        float bv = bias ? bias[col] : 0.f;
#pragma unroll
        for (int v = 0; v < 8; v++) {
            int m = m0 + v + 8 * hi;
            C[(size_t)m * N + col] = acc[j][v] + bv;
        }
    }
}

// ---------------------------------------------------------------------------
// One LSTM time step, both directions. grid=(64 h-tiles, 2 dirs), block=128.
// Wave 0 issues a 1-D TENSOR_LOAD_TO_LDS of the full h_in [16x1024] (64 KB)
// into LDS (TDM D# per ISA 8.3/8.4: count=1, type=2, data_size=4B,
// tensor_dim0=tile_dim0=16384), waits TENSORcnt==0, then all waves barrier.
// Wave w computes gate quadrant w (i/f/g/o) 16x16 WMMA tile over K=1024 of
// h_lds @ Wh, initialized from the xW slice (bias folded in at GEMM time).
// Gates -> LDS -> fused sigmoid/tanh -> c (tile-private, in place),
// h_out (ping-pong), hs output.
// ---------------------------------------------------------------------------
__global__ void lstm_step(const float* __restrict__ xWu, const float* __restrict__ Whu,
                          const float* __restrict__ hui, float* __restrict__ huo,
                          float* __restrict__ cu, float* __restrict__ upo, int tu,
                          const float* __restrict__ xWd, const float* __restrict__ Whd,
                          const float* __restrict__ hdi, float* __restrict__ hdo,
                          float* __restrict__ cd, float* __restrict__ dno, int td) {
    const float *xW, *Wh, *hin;
    float *hout, *cio, *hs;
    int t;
    if (blockIdx.y == 0) { xW = xWu; Wh = Whu; hin = hui; hout = huo; cio = cu; hs = upo; t = tu; }
    else                 { xW = xWd; Wh = Whd; hin = hdi; hout = hdo; cio = cd; hs = dno; t = td; }

    __shared__ float hsh[Bz * RNN];      // 64 KB staged h_in (TDM destination)
    __shared__ float gts[4][16][16];     // 4 KB gate tiles

    if ((threadIdx.x >> 5) == 0) {       // wave 0 issues the TDM descriptor
        unsigned long long ga = (unsigned long long)(uintptr_t)hin;   // byte addr
        unsigned int ldsb = (unsigned int)(uintptr_t)(void*)hsh;      // flat->LDS: addr[31:0]
        u32x4 g0;
        g0.x = 1u;                                            // count=1, user mode
        g0.y = ldsb;                                          // lds_addr (bytes)
        g0.z = (unsigned int)ga;                              // global_addr[31:0]
        g0.w = (unsigned int)((ga >> 32) & 0x01FFFFFFu)       // global_addr[56:32]
             | (2u << 30);                                    // type=2 ("image")
        u32x8 g1;
        g1[0] = (2u << 16);          // workgroup_mask=0 (no cluster), data_size=4B
        g1[1] = (16384u & 0xFFFFu) << 16;   // tensor_dim0[15:0] = 16384 elements
        g1[2] = 0;                          // tensor_dim0[31:16]=0, tensor_dim1 lo=0
        g1[3] = (16384u << 16);             // tile_dim0 = 16384 (1-D tile)
        g1[4] = 0;                          // tile_dim1=0 (unused), tile_dim2=0
        g1[5] = 16384u;                     // tensor_dim0_stride lo
        g1[6] = 0;
        g1[7] = 0;
        asm volatile("tensor_load_to_lds %0, %1"
                     :: "s"(g0), "s"(g1)
                     : "memory");
        __builtin_amdgcn_s_wait_tensorcnt(0);
    }
    __syncthreads();                      // LDS h now visible to all 4 waves

    int w = threadIdx.x >> 5;            // gate quadrant 0..3
    int lane = threadIdx.x & 31;
    int l = lane & 15, hi = lane >> 4;
    int n0 = blockIdx.x * 16;            // h column tile
    int gcol = w * RNN + n0 + l;         // gate column in [0, 4096)

    v8f acc;
#pragma unroll
    for (int v = 0; v < 8; v++)
        acc[v] = xW[((size_t)(v + 8 * hi) * Tz + t) * (4 * RNN) + gcol];

    const float* hrow = hsh + (size_t)l * RNN;   // A-fragments from LDS
    for (int kk = 0; kk < RNN; kk += 4) {
        int ka = kk + 2 * hi;
        v2f a = *(const v2f*)(hrow + ka);        // ds_load_b64
        v2f b;
        b.x = Wh[(size_t)ka * (4 * RNN) + gcol];
        b.y = Wh[((size_t)ka + 1) * (4 * RNN) + gcol];
        acc = __builtin_amdgcn_wmma_f32_16x16x4_f32(
            false, a, false, b, (short)0, acc, false, false);
    }

#pragma unroll
    for (int v = 0; v < 8; v++) gts[w][v + 8 * hi][l] = acc[v];
    __syncthreads();

    for (int idx = threadIdx.x; idx < 256; idx += 128) {
        int m = idx >> 4, j = idx & 15;
        float iv = gts[0][m][j], fv = gts[1][m][j];
        float gv = gts[2][m][j], ov = gts[3][m][j];
        size_t ci = (size_t)m * RNN + n0 + j;
        float cp = cio[ci];
        float cn = sigf(fv) * cp + sigf(iv) * tanhf(gv);
        float hn = sigf(ov) * tanhf(cn);
        cio[ci] = cn;
        hout[ci] = hn;
        hs[((size_t)m * Tz + t) * RNN + n0 + j] = hn;
    }
}

// lg[b,t,ch] = concat(up,dn)[b,t,:] . W_attn[:,ch] + b_attn[ch]; wave per (b,t)
__global__ void attn_logits(const float* __restrict__ up, const float* __restrict__ dn,
                            const float* __restrict__ Wa, const float* __restrict__ ba,
                            float* __restrict__ lg) {
    int wid = blockIdx.x * (blockDim.x >> 5) + (threadIdx.x >> 5);
    int lane = threadIdx.x & 31;
    if (wid >= Bz * Tz) return;
    const float* u = up + (size_t)wid * RNN;
    const float* d = dn + (size_t)wid * RNN;
    float a0 = 0.f, a1 = 0.f, a2 = 0.f;
    for (int k = lane; k < 2 * RNN; k += 32) {
        float hv = (k < RNN) ? u[k] : d[k - RNN];
        const float* wr = Wa + (size_t)k * 3;
        a0 += hv * wr[0]; a1 += hv * wr[1]; a2 += hv * wr[2];
    }
    a0 = wred(a0); a1 = wred(a1); a2 = wred(a2);
    if (lane == 0) {
        lg[wid * 3 + 0] = a0 + ba[0];
        lg[wid * 3 + 1] = a1 + ba[1];
        lg[wid * 3 + 2] = a2 + ba[2];
    }
}

// Causal prefix softmax over s<=t of lg[:,ch], weighted sum of Q rows.
// grid=(B*T, 3 channels), block=320 (thread = embed column).
__global__ void prefix_embed(const float* __restrict__ lg, const float* __restrict__ Q,
                             float* __restrict__ E3) {
    int bt = blockIdx.x, ch = blockIdx.y;
    int b = bt >> 6, t = bt & 63;
    int col = threadIdx.x;
    float mx = -3.402823e38f;
    for (int s = 0; s <= t; s++)
        mx = fmaxf(mx, lg[((b << 6) + s) * 3 + ch]);
    float wsum = 0.f, accv = 0.f;
    for (int s = 0; s <= t; s++) {
        float w = expf(lg[((b << 6) + s) * 3 + ch] - mx);
        wsum += w;
        if (col < EMBED) accv += w * Q[(size_t)((b << 6) + s) * EMBED + col];
    }
    if (col < EMBED)
        E3[(size_t)ch * (Bz * Tz * EMBED) + (size_t)bt * EMBED + col] = accv / wsum;
}

// Fused scoring tail. Wave per (b,t); lane holds 8 channels; obj score
// reduced+broadcast (xor butterfly) before pair score. b_s folded into es/eo,
// b_p folded into er at GEMM time.
__global__ void final_score(const float* __restrict__ es, const float* __restrict__ eo,
                            const float* __restrict__ er,
                            const float* __restrict__ vps, const float* __restrict__ vpp,
                            const float* __restrict__ w_s, const float* __restrict__ w_o,
                            const float* __restrict__ w_p,
                            const float* __restrict__ mask, float* __restrict__ out) {
    int wid = blockIdx.x * (blockDim.x >> 5) + (threadIdx.x >> 5);
    int lane = threadIdx.x & 31;
    if (wid >= Bz * Tz) return;
    int b = wid >> 6;

    float esr[8], eor[8], err[8], wsr[8], wor[8], wpr[8];
#pragma unroll
    for (int u = 0; u < 8; u++) {
        int c = lane + 32 * u;
        esr[u] = es[(size_t)wid * SCd + c];
        eor[u] = eo[(size_t)wid * SCd + c];
        err[u] = er[(size_t)wid * SCd + c];
        wsr[u] = w_s[c]; wor[u] = w_o[c]; wpr[u] = w_p[c];
    }

    for (int r = 0; r < Rz; r++) {
        const float* vs = vps + (size_t)(b * Rz + r) * SCd;
        const float* vp = vpp + (size_t)(b * Rz + r) * SCd;
        float sbj = 0.f, obj = 0.f;
        float vpv[8];
#pragma unroll
        for (int u = 0; u < 8; u++) {
            int c = lane + 32 * u;
            float vsv = vs[c];
            vpv[u] = vp[c];
            sbj += tanhf(esr[u] + vsv) * wsr[u];
            obj += tanhf(eor[u] + vsv) * wsr[u];
        }
        sbj = wred(sbj);
        obj = wred(obj);   // broadcast to all lanes
        float pair = 0.f;
#pragma unroll
        for (int u = 0; u < 8; u++)
            pair += tanhf(err[u] + vpv[u] + obj * wor[u]) * wpr[u];
        pair = wred(pair);
        if (lane == 0)
            out[(size_t)wid * Rz + r] = (sbj + pair) * (1.0f / 32.0f)
                                        + mask[(size_t)wid * Rz + r];
    }
}

// ---------------------------------------------------------------------------

extern "C" void kernel_launch(void* const* d_in, const int* in_sizes, int n_in,
                              void* d_out, int out_size, void* d_ws, size_t ws_size,
                              hipStream_t stream) {
    (void)in_sizes; (void)n_in; (void)out_size; (void)ws_size;
    const float* encT  = (const float*)d_in[0];
    const float* encI  = (const float*)d_in[1];
    const float* mask  = (const float*)d_in[2];
    const float* WQ    = (const float*)d_in[3];
    const float* bQ    = (const float*)d_in[4];
    const float* WK    = (const float*)d_in[5];
    const float* bK    = (const float*)d_in[6];
    const float* Wx_up = (const float*)d_in[7];
    const float* Wh_up = (const float*)d_in[8];
    const float* b_up  = (const float*)d_in[9];
    const float* Wx_dn = (const float*)d_in[10];
    const float* Wh_dn = (const float*)d_in[11];
    const float* b_dn  = (const float*)d_in[12];
    const float* W_at  = (const float*)d_in[13];
    const float* b_at  = (const float*)d_in[14];
    const float* Wv_s  = (const float*)d_in[15];
    const float* We_s  = (const float*)d_in[16];
    const float* b_s   = (const float*)d_in[17];
    const float* w_s   = (const float*)d_in[18];
    const float* Wv_p  = (const float*)d_in[19];
    const float* We_p  = (const float*)d_in[20];
    const float* b_p   = (const float*)d_in[21];
    const float* w_o   = (const float*)d_in[22];
    const float* w_p   = (const float*)d_in[23];
    float* out = (float*)d_out;

    // -------- workspace partition (floats), ~60 MB total --------
    float* ws = (float*)d_ws;
    size_t off = 0;
    auto take = [&](size_t n) { float* p = ws + off; off += n; return p; };
    float* Qb   = take((size_t)Bz * Tz * EMBED);        // 307,200
    float* Kb   = take((size_t)Bz * Rz * RNN);          // 1,638,400
    float* xWu  = take((size_t)Bz * Tz * 4 * RNN);      // 4,194,304
    float* xWd  = take((size_t)Bz * Tz * 4 * RNN);
    float* up   = take((size_t)Bz * Tz * RNN);          // 1,048,576
    float* dn   = take((size_t)Bz * Tz * RNN);
    float* E3   = take((size_t)3 * Bz * Tz * EMBED);    // sbj/rlt/obj embeds
    float* es   = take((size_t)Bz * Tz * SCd);
    float* eo   = take((size_t)Bz * Tz * SCd);
    float* er   = take((size_t)Bz * Tz * SCd);
    float* vps  = take((size_t)Bz * Rz * SCd);
    float* vpp  = take((size_t)Bz * Rz * SCd);
    float* lg   = take((size_t)Bz * Tz * 3);
    float* hupA = take((size_t)Bz * RNN);   // zeroed group start
    float* hdnA = take((size_t)Bz * RNN);
    float* cup  = take((size_t)Bz * RNN);
    float* cdn  = take((size_t)Bz * RNN);
    float* hupB = take((size_t)Bz * RNN);
    float* hdnB = take((size_t)Bz * RNN);

    dim3 blk32(32);

    // zero initial h (A buffers) and c — contiguous 4*16K floats
    zero_kernel<<<(4 * Bz * RNN + 255) / 256, 256, 0, stream>>>(hupA, 4 * Bz * RNN);

    // Q = encT @ WQ + bQ   [1024,768]x[768,300]  (N tail)
    gemm_wmma<1, true><<<dim3(64, (EMBED + 15) / 16), blk32, 0, stream>>>(
        encT, WQ, bQ, Qb, Bz * Tz, EMBED, 768);
    // K = encI @ WK + bK   [1600,2048]x[2048,1024]
    gemm_wmma<4, false><<<dim3(100, 16), blk32, 0, stream>>>(
        encI, WK, bK, Kb, Bz * Rz, RNN, 2048);
    // xW = Q @ Wx + b      [1024,300]x[300,4096]
    gemm_wmma<4, false><<<dim3(64, 64), blk32, 0, stream>>>(
        Qb, Wx_up, b_up, xWu, Bz * Tz, 4 * RNN, EMBED);
    gemm_wmma<4, false><<<dim3(64, 64), blk32, 0, stream>>>(
        Qb, Wx_dn, b_dn, xWd, Bz * Tz, 4 * RNN, EMBED);

    // serial bidirectional scan: up t=s, down t=63-s; ping-pong h
    for (int s = 0; s < Tz; s++) {
        float* hui = (s & 1) ? hupB : hupA;
        float* huo = (s & 1) ? hupA : hupB;
        float* hdi = (s & 1) ? hdnB : hdnA;
        float* hdo = (s & 1) ? hdnA : hdnB;
        lstm_step<<<dim3(RNN / 16, 2), 128, 0, stream>>>(
            xWu, Wh_up, hui, huo, cup, up, s,
            xWd, Wh_dn, hdi, hdo, cdn, dn, Tz - 1 - s);
    }

    attn_logits<<<256, 128, 0, stream>>>(up, dn, W_at, b_at, lg);
    prefix_embed<<<dim3(Bz * Tz, 3), 320, 0, stream>>>(lg, Qb, E3);

    const size_t Estride = (size_t)Bz * Tz * EMBED;
    // es = e_sbj@We_s + b_s ; eo = e_obj@We_s + b_s ; er = e_rlt@We_p + b_p
    gemm_wmma<4, false><<<dim3(64, 4), blk32, 0, stream>>>(
        E3 + 0 * Estride, We_s, b_s, es, Bz * Tz, SCd, EMBED);
    gemm_wmma<4, false><<<dim3(64, 4), blk32, 0, stream>>>(
        E3 + 2 * Estride, We_s, b_s, eo, Bz * Tz, SCd, EMBED);
    gemm_wmma<4, false><<<dim3(64, 4), blk32, 0, stream>>>(
        E3 + 1 * Estride, We_p, b_p, er, Bz * Tz, SCd, EMBED);
    // vp_s = K@Wv_s ; vp_p = K@Wv_p (no bias)
    gemm_wmma<4, false><<<dim3(100, 4), blk32, 0, stream>>>(
        Kb, Wv_s, nullptr, vps, Bz * Rz, SCd, RNN);
    gemm_wmma<4, false><<<dim3(100, 4), blk32, 0, stream>>>(
        Kb, Wv_p, nullptr, vpp, Bz * Rz, SCd, RNN);

    final_score<<<256, 128, 0, stream>>>(es, eo, er, vps, vpp,
                                         w_s, w_o, w_p, mask, out);
}